// GazePreprocessor_24232205484643
// MI455X (gfx1250) — compile-verified
//
#include <hip/hip_runtime.h>

typedef __attribute__((ext_vector_type(2))) float v2f;
typedef __attribute__((ext_vector_type(8))) float v8f;

#define IMG_H  180
#define IMG_W  320
#define BAND_LO 60      // nonzero output rows: [60, 120)
#define BAND_H  60
#define KSIZE  121      // gaussian taps, valid index range [0,120]
#define MAXP   5
#define N_IMG  1024     // B*T = 16*64
#define PSTRIDE 10      // padded K-row stride (floats): b64-aligned, conflict-free

// One workgroup per image, 8 waves (256 threads, wave32).
// Band heat[60x320] = A[60x8] x B[8x320] via V_WMMA_F32_16X16X4_F32 (K 5->8).
// A/B operand tables built once in LDS (point-major, padded rows) so each
// fragment is a single ds_load_b64. Fused max-normalize + b128 writes.
__global__ __launch_bounds__(256)
void gaze_heatmap_wmma(const float* __restrict__ gaze,
                       const float* __restrict__ kern,
                       float* __restrict__ out)
{
    __shared__ float s_kern[128];                      // taps, [121..127] = 0
    __shared__ __align__(16) float s_heat[64 * IMG_W]; // 64 rows: unmasked tile stores
    __shared__ __align__(16) float s_amat[64 * PSTRIDE];   // A[m][p], p-major
    __shared__ __align__(16) float s_bmat[IMG_W * PSTRIDE];// B[x][p], p-major
    __shared__ float s_w[8];
    __shared__ int   s_px[8], s_py[8];
    __shared__ float s_red[256];

    const int tid = threadIdx.x;
    const int img = blockIdx.x;

    __builtin_prefetch(gaze + (size_t)img * (MAXP * 2), 0, 3);  // global_prefetch_b8
    __builtin_prefetch(kern, 0, 3);

    // --- stage 0: gaussian taps -> LDS (zero-padded) ----------------------
    if (tid < 128) s_kern[tid] = (tid < KSIZE) ? kern[tid] : 0.0f;

    // --- stage 1: per-image point decode (5 points, serial on thread 0) ---
    if (tid == 0) {
        const float* g = gaze + (size_t)img * (MAXP * 2);
        float wrun = 1.0f;                   // COEFF^rank over valid points
        for (int p = 0; p < MAXP; ++p) {
            float cx = g[2 * p + 0], cy = g[2 * p + 1];
            bool valid = (cx >= 0.0f) && (cy >= 0.0f);
            float xc = fminf(fmaxf(cx, 0.0f), 1.0f) * (float)(IMG_W - 1);
            float yc = fminf(fmaxf(cy, 0.0f), 1.0f) * (float)(IMG_H - 1);
            int xi = (int)xc; xi = min(max(xi, 0), IMG_W - 1);
            int yi = (int)yc; yi = min(max(yi, 0), IMG_H - 1);
            s_px[p] = xi; s_py[p] = yi;
            if (valid) { s_w[p] = wrun; wrun *= 0.8f; } else { s_w[p] = 0.0f; }
        }
        for (int p = MAXP; p < 8; ++p) { s_w[p] = 0.0f; s_px[p] = 0; s_py[p] = 0; }
    }
    __syncthreads();

    // Branch-free tap: safe index 127 holds 0.0f
    #define TAP(i) s_kern[((unsigned)(i) <= 120u) ? (i) : 127]

    // --- stage 2a: build operand tables (once per image) ------------------
    // A[m][p] = w_p * k[y_p - m]        (64 x 8, 512 lookups)
    for (int e = tid; e < 64 * 8; e += 256) {
        const int m = e >> 3, p = e & 7;
        s_amat[m * PSTRIDE + p] = s_w[p] * TAP(s_py[p] - m);
    }
    // B[x][p] = k[x_p - x + 60]         (320 x 8, 2560 lookups)
    for (int e = tid; e < IMG_W * 8; e += 256) {
        const int x = e >> 3, p = e & 7;
        s_bmat[x * PSTRIDE + p] = TAP(s_px[p] - x + 60);
    }
    #undef TAP
    __syncthreads();

    // --- stage 2b: band GEMM via WMMA f32 16x16x4 -------------------------
    // Wave w owns M-tile mt = w>>1 and N-tiles (w&1)*10 .. (w&1)*10+9.
    const int wave  = tid >> 5;
    const int lane  = tid & 31;
    const int r16   = lane & 15;   // M position (A) / N position (B) in tile
    const int khalf = lane >> 4;   // lanes 16-31 hold K={2,3} of each K-quad
    const int m0    = (wave >> 1) * 16;
    const int nhalf = wave & 1;
    const int p0    = khalf * 2;   // this lane's first K slot

    // A fragments: one b64 each, hoisted per wave (fixed M-tile)
    const float* arow = &s_amat[(m0 + r16) * PSTRIDE + p0];
    const v2f a_lo = *(const v2f*)(arow);       // K slots p0, p0+1
    const v2f a_hi = *(const v2f*)(arow + 4);   // K slots p0+4, p0+5

    for (int j = 0; j < 10; ++j) {
        const int n0 = (nhalf * 10 + j) * 16;
        const float* brow = &s_bmat[(n0 + r16) * PSTRIDE + p0];
        const v2f b_lo = *(const v2f*)(brow);
        const v2f b_hi = *(const v2f*)(brow + 4);

        v8f c = {0.f, 0.f, 0.f, 0.f, 0.f, 0.f, 0.f, 0.f};
        c = __builtin_amdgcn_wmma_f32_16x16x4_f32(
                false, a_lo, false, b_lo, (short)0, c, false, false);
        c = __builtin_amdgcn_wmma_f32_16x16x4_f32(
                false, a_hi, false, b_hi, (short)0, c, false, false);

        // D layout: lane = N + 16*(M>=8), vgpr r -> M = 8*khalf + r
        #pragma unroll
        for (int r = 0; r < 8; ++r) {
            const int m = m0 + khalf * 8 + r;
            s_heat[m * IMG_W + (n0 + r16)] = c[r];
        }
    }
    __syncthreads();

    // --- stage 3: max over the 60x320 band --------------------------------
    float lmax = 0.0f;
    for (int i = tid; i < BAND_H * IMG_W; i += 256)
        lmax = fmaxf(lmax, s_heat[i]);
    s_red[tid] = lmax;
    __syncthreads();
    #pragma unroll
    for (int s = 128; s > 0; s >>= 1) {
        if (tid < s) s_red[tid] = fmaxf(s_red[tid], s_red[tid + s]);
        __syncthreads();
    }
    const float inv = 1.0f / (s_red[0] + 1e-8f);

    // --- stage 4: single-pass normalized write (b128), zero outside band --
    float4* out4 = (float4*)(out + (size_t)img * (IMG_H * IMG_W));
    for (int q = tid; q < (IMG_H * IMG_W) / 4; q += 256) {
        const int pix = q * 4;
        const int y = pix / IMG_W;
        const int x = pix - y * IMG_W;          // IMG_W % 4 == 0: no row crossing
        float4 v = {0.0f, 0.0f, 0.0f, 0.0f};
        if (y >= BAND_LO && y < BAND_LO + BAND_H) {
            const int base = (y - BAND_LO) * IMG_W + x;
            v.x = s_heat[base + 0] * inv;
            v.y = s_heat[base + 1] * inv;
            v.z = s_heat[base + 2] * inv;
            v.w = s_heat[base + 3] * inv;
        }
        out4[q] = v;
    }
}

extern "C" void kernel_launch(void* const* d_in, const int* in_sizes, int n_in,
                              void* d_out, int out_size, void* d_ws, size_t ws_size,
                              hipStream_t stream) {
    const float* gaze = (const float*)d_in[0];   // [16,64,10] f32
    const float* kern = (const float*)d_in[1];   // [121] f32
    float* out = (float*)d_out;                  // [16,64,1,180,320] f32
    (void)in_sizes; (void)n_in; (void)out_size; (void)d_ws; (void)ws_size;
    gaze_heatmap_wmma<<<N_IMG, 256, 0, stream>>>(gaze, kern, out);
}